// PointPillarScatter_41515153883554
// MI455X (gfx1250) — compile-verified
//
#include <hip/hip_runtime.h>

// PointPillarScatter for MI455X (gfx1250, wave32).
// B=4, C=64 channels, P=12000 pillars, grid G = 496*432 = 214272.
// out[b][c][g] = feats[b][c][p] where pillar p of batch b maps to cell g
// (coords[i] = (b, g), mask g>0), else 0.
//
// Strategy: invert scatter -> gather so the 219MB output is ONE streaming
// non-temporal write pass; the 12.3MB feature array and 3.4MB inverse map
// stay L2-resident (192MB L2), so gathers are cheap. Map chunks are staged
// to LDS with gfx1250 async-to-LDS ops (ASYNCcnt).

#define C_NBF 64
#define C_NX  432
#define C_NY  496
#define C_G   (C_NX * C_NY)   // 214272  (NZ==1)
#define C_B   4
#define C_P   12000
#define C_CSPLIT 4            // channel groups per (b, g-chunk)
#define C_CPB (C_NBF / C_CSPLIT)  // 16 channels per block

// Native clang vector types: required by __builtin_nontemporal_store and
// guarantee single b128 vmem ops.
typedef float f32x4 __attribute__((ext_vector_type(4)));
typedef int   i32x4 __attribute__((ext_vector_type(4)));

// Exact parameter type of __builtin_amdgcn_global_load_async_to_lds_b128
// (per hipcc diagnostic): pointer to gcc-style int vector, with address space.
typedef int v4i_vs __attribute__((vector_size(16)));
typedef __attribute__((address_space(1))) v4i_vs as1_v4i;
typedef __attribute__((address_space(3))) v4i_vs as3_v4i;

#if defined(__has_builtin)
#if __has_builtin(__builtin_amdgcn_global_load_async_to_lds_b128)
#define PPS_HAVE_ASYNC_LDS 1
#endif
#endif

// ---- pass 0: fill inverse map with -1 (empty) -------------------------------
__global__ __launch_bounds__(256) void pps_init_map(i32x4* __restrict__ map4) {
    const int t = blockIdx.x * 256 + threadIdx.x;
    if (t < (C_B * C_G) / 4) {
        map4[t] = (i32x4)(-1);   // splat; regular store: we want it in L2
    }
}

// ---- pass 1: scatter pillar id into map -------------------------------------
// coords row i = (batch, spatial); pillar i corresponds to feats row i = b*P+p.
__global__ __launch_bounds__(256) void pps_scatter_idx(const int* __restrict__ coords,
                                                       int* __restrict__ map) {
    const int i = blockIdx.x * 256 + threadIdx.x;
    if (i < C_B * C_P) {
        const int b = coords[2 * i];
        const int g = coords[2 * i + 1];
        if (g > 0) {
            map[b * C_G + g] = i;   // unique per (b,g): perm is a permutation
        }
    }
}

// ---- pass 2: streaming gather + non-temporal write of the full output -------
// Block owns 1024 consecutive g cells (4 per thread) and 16 channels.
// Map chunk (4KB) is staged to LDS via gfx1250 async-to-LDS, then each output
// store is a 16B TH=NT store so the 219MB stream doesn't evict feats/map
// from L2. feats index: (b*C + c)*P + (m - b*P) = b*(C-1)*P + c*P + m.
__global__ __launch_bounds__(256) void pps_gather_out(const float* __restrict__ feats,
                                                      const int* __restrict__ map,
                                                      float* __restrict__ out) {
    __shared__ int smap[1024];                       // 256 threads * int4

    const int b  = blockIdx.y;
    const int c0 = blockIdx.z * C_CPB;
    const int g0 = (blockIdx.x * 256 + threadIdx.x) * 4;
    if (g0 >= C_G) return;

    const int* __restrict__ gsrc = map + (size_t)b * C_G + g0;

    i32x4 m;
#ifdef PPS_HAVE_ASYNC_LDS
    {
        // async global->LDS copy of this lane's 16B map slice (ASYNCcnt-tracked)
        v4i_vs* gsrc_v = (v4i_vs*)gsrc;                       // pointee reinterpret
        v4i_vs* lds_v  = (v4i_vs*)&smap[threadIdx.x * 4];
        __builtin_amdgcn_global_load_async_to_lds_b128(
            (as1_v4i*)gsrc_v, (as3_v4i*)lds_v, 0, 0);
#if __has_builtin(__builtin_amdgcn_s_wait_asynccnt)
        __builtin_amdgcn_s_wait_asynccnt(0);
#else
        asm volatile("s_wait_asynccnt 0x0" ::: "memory");
#endif
        m = *reinterpret_cast<const i32x4*>(&smap[threadIdx.x * 4]);  // ds_load_b128
    }
#else
    (void)smap;
    m = *reinterpret_cast<const i32x4*>(gsrc);
#endif

    const float* __restrict__ base = feats + (size_t)b * (C_NBF - 1) * C_P;
    float* __restrict__ ob = out + ((size_t)(b * C_NBF + c0)) * C_G + g0;

    #pragma unroll 4
    for (int c = c0; c < c0 + C_CPB; ++c) {
        const float* __restrict__ fc = base + (size_t)c * C_P;
        f32x4 v;
        v.x = (m.x >= 0) ? fc[m.x] : 0.0f;
        v.y = (m.y >= 0) ? fc[m.y] : 0.0f;
        v.z = (m.z >= 0) ? fc[m.z] : 0.0f;
        v.w = (m.w >= 0) ? fc[m.w] : 0.0f;
        // warm next channel's cacheline for one gather stream (global_prefetch_b8)
        if ((c + 1 < c0 + C_CPB) && (m.x >= 0)) {
            __builtin_prefetch(fc + C_P + m.x, 0, 1);
        }
        // TH=NT store: keep the 219MB output stream out of L2's working set
        __builtin_nontemporal_store(v, reinterpret_cast<f32x4*>(ob));
        ob += (size_t)C_G;
    }
}

// ---- fallback path (only if workspace is too small for the map) -------------
__global__ __launch_bounds__(256) void pps_zero_out(f32x4* __restrict__ out4, int n4) {
    const int t = blockIdx.x * 256 + threadIdx.x;
    if (t < n4) {
        __builtin_nontemporal_store((f32x4)(0.0f), out4 + t);
    }
}

__global__ __launch_bounds__(256) void pps_scatter_direct(const float* __restrict__ feats,
                                                          const int* __restrict__ coords,
                                                          float* __restrict__ out) {
    const int t = blockIdx.x * 256 + threadIdx.x;
    if (t >= C_B * C_P * C_NBF) return;
    const int c = t & (C_NBF - 1);
    const int i = t >> 6;
    const int b = coords[2 * i];
    const int g = coords[2 * i + 1];
    if (g > 0) {
        const int p = i - b * C_P;
        out[((size_t)(b * C_NBF + c)) * C_G + g] =
            feats[((size_t)(b * C_NBF + c)) * C_P + p];
    }
}

extern "C" void kernel_launch(void* const* d_in, const int* in_sizes, int n_in,
                              void* d_out, int out_size, void* d_ws, size_t ws_size,
                              hipStream_t stream) {
    const float* feats  = (const float*)d_in[0];   // (4, 64, 12000, 1) f32
    const int*   coords = (const int*)d_in[1];     // (48000, 2) int
    float*       out    = (float*)d_out;           // (4, 64, 496, 432) f32
    (void)in_sizes; (void)n_in;

    const size_t map_bytes = (size_t)C_B * C_G * sizeof(int);  // ~3.4 MB

    if (d_ws != nullptr && ws_size >= map_bytes) {
        int* map = (int*)d_ws;

        const int n_init = (C_B * C_G) / 4;                    // 214272 int4s
        pps_init_map<<<(n_init + 255) / 256, 256, 0, stream>>>((i32x4*)map);

        const int n_pil = C_B * C_P;                           // 48000
        pps_scatter_idx<<<(n_pil + 255) / 256, 256, 0, stream>>>(coords, map);

        dim3 grid((C_G / 4 + 255) / 256, C_B, C_CSPLIT);       // (210, 4, 4)
        pps_gather_out<<<grid, 256, 0, stream>>>(feats, map, out);
    } else {
        const int n4 = out_size / 4;                           // 13713408
        pps_zero_out<<<(n4 + 255) / 256, 256, 0, stream>>>((f32x4*)out, n4);

        const int n = C_B * C_P * C_NBF;                       // 3072000
        pps_scatter_direct<<<(n + 255) / 256, 256, 0, stream>>>(feats, coords, out);
    }
}